// LFE_34600256537190
// MI455X (gfx1250) — compile-verified
//
#include <hip/hip_runtime.h>
#include <hip/hip_bf16.h>

// Problem constants (from reference): B=4, C=64, N=40960, K=16, D=128
#define BB 4
#define CC 64
#define NN 40960
#define KK 16
#define DD 128

typedef float v2f __attribute__((ext_vector_type(2)));
typedef float v8f __attribute__((ext_vector_type(8)));

// ---------------------------------------------------------------------------
// Kernel 0: precompute BN scale/shift:  scale = g/sqrt(v+eps), shift = b-m*scale
// P layout (floats): [0..63]=sc1 [64..127]=sh1 [128..255]=sc2 [256..383]=sh2
//                    [384..511]=sc3 [512..639]=sh3
// ---------------------------------------------------------------------------
__global__ __launch_bounds__(128) void bn_prep(
    const float* __restrict__ g1, const float* __restrict__ b1,
    const float* __restrict__ m1, const float* __restrict__ v1,
    const float* __restrict__ g2, const float* __restrict__ b2,
    const float* __restrict__ m2, const float* __restrict__ v2,
    const float* __restrict__ g3, const float* __restrict__ b3,
    const float* __restrict__ m3, const float* __restrict__ v3,
    float* __restrict__ P)
{
    const int t = threadIdx.x;      // 0..127
    if (t < CC) {
        float s = g1[t] / sqrtf(v1[t] + 1e-5f);
        P[t]      = s;
        P[64 + t] = b1[t] - m1[t] * s;
    }
    float s2 = g2[t] / sqrtf(v2[t] + 1e-5f);
    P[128 + t] = s2;
    P[256 + t] = b2[t] - m2[t] * s2;
    float s3 = g3[t] / sqrtf(v3[t] + 1e-5f);
    P[384 + t] = s3;
    P[512 + t] = b3[t] - m3[t] * s3;
}

// ---------------------------------------------------------------------------
// Kernel 1: t1[b][n][o] = relu(sc1[o] * (w1 @ f)[b,o,n] + sh1[o])
// f layout: (B, C, N).  t1 layout: (B, N, 64) -> gathered columns contiguous.
// Workgroup = 256 threads = 8 waves; each wave computes a full 64x16 tile
// (all 64 output channels for 16 points) via 4 m-tiles x 16 k-steps of
// v_wmma_f32_16x16x4_f32.
// ---------------------------------------------------------------------------
__global__ __launch_bounds__(256) void stage1(
    const float* __restrict__ f, const float* __restrict__ w1,
    const float* __restrict__ P, float* __restrict__ t1)
{
    const int lane = threadIdx.x & 31;
    const int wave = threadIdx.x >> 5;
    const int ln   = lane & 15;     // M (for A) / N (for B) position
    const int hh   = lane >> 4;     // half: selects K pair

    const long long gp = (long long)blockIdx.x * 128 + wave * 16;
    const int b  = (int)(gp / NN);
    const int n0 = (int)(gp % NN);

    const float* fb = f + (long long)b * CC * NN;
    const float* sc1 = P;
    const float* sh1 = P + 64;

    v8f acc[4] = {v8f{}, v8f{}, v8f{}, v8f{}};

    #pragma unroll
    for (int k = 0; k < CC; k += 4) {
        const int kk = k + 2 * hh;
        // B tile: 4x16 slice of f  (B[k][n] = f[b, k, n0+n])
        v2f bt;
        bt.x = fb[(long long)kk * NN + n0 + ln];
        bt.y = fb[(long long)(kk + 1) * NN + n0 + ln];
        #pragma unroll
        for (int mt = 0; mt < 4; ++mt) {
            const int m = mt * 16 + ln;
            v2f at;
            at.x = w1[m * CC + kk];
            at.y = w1[m * CC + kk + 1];
            acc[mt] = __builtin_amdgcn_wmma_f32_16x16x4_f32(
                false, at, false, bt, (short)0, acc[mt], false, false);
        }
    }

    // BN + ReLU, store to (b, n, o) layout: element (mt,r,lane) has
    // o = mt*16 + r + 8*hh, n = n0 + ln
    float* dst = t1 + ((long long)b * NN + n0 + ln) * 64;
    #pragma unroll
    for (int mt = 0; mt < 4; ++mt) {
        #pragma unroll
        for (int r = 0; r < 8; ++r) {
            const int o = mt * 16 + r + 8 * hh;
            float x = acc[mt][r] * sc1[o] + sh1[o];
            dst[o] = fmaxf(x, 0.0f);
        }
    }
}

// ---------------------------------------------------------------------------
// Kernel 2: per 64-point tile:
//   hs[p][c] = sum_k t1[b][idx[b,n,k]][c]                (gather-reduce, LDS)
//   out[b,o,n] = relu(sc2*(w2@hs) + sh2) + relu(sc3*(w3@f) + sh3)
// LDS hs padded to stride 68 floats -> conflict-free B-tile reads.
// 8 waves: wave/2 selects 16-point tile, wave&1 selects output-channel half.
// ---------------------------------------------------------------------------
__global__ __launch_bounds__(256) void stage2(
    const float* __restrict__ f, const int* __restrict__ nidx,
    const float* __restrict__ t1,
    const float* __restrict__ w2, const float* __restrict__ w3,
    const float* __restrict__ P, float* __restrict__ out)
{
    __shared__ float hs[64][68];

    const long long gp0 = (long long)blockIdx.x * 64;
    const int b  = (int)(gp0 / NN);
    const int n0 = (int)(gp0 % NN);

    // ---- Phase 1: gather-reduce neighbors into LDS (4 threads per point) ----
    {
        const int p = threadIdx.x >> 2;   // point within tile 0..63
        const int q = threadIdx.x & 3;    // 16-channel block 0..3
        const int n = n0 + p;
        const int* ip = nidx + ((long long)b * NN + n) * KK;

        float a[16];
        #pragma unroll
        for (int i = 0; i < 16; ++i) a[i] = 0.0f;

        for (int k = 0; k < KK; ++k) {
            const int j = ip[k];
            const float4* col =
                (const float4*)(t1 + ((long long)b * NN + j) * 64 + q * 16);
            #pragma unroll
            for (int v = 0; v < 4; ++v) {
                float4 c = col[v];
                a[4 * v + 0] += c.x;
                a[4 * v + 1] += c.y;
                a[4 * v + 2] += c.z;
                a[4 * v + 3] += c.w;
            }
        }
        float* drow = &hs[p][q * 16];
        #pragma unroll
        for (int i = 0; i < 16; ++i) drow[i] = a[i];
    }
    __syncthreads();

    // ---- Phase 2: two WMMA GEMMs + fused BN/ReLU/add, coalesced store ----
    const int lane = threadIdx.x & 31;
    const int wave = threadIdx.x >> 5;
    const int pt = wave >> 1;          // 16-point tile 0..3
    const int mh = wave & 1;           // output-channel half (0 or 1)
    const int ln = lane & 15;
    const int hh = lane >> 4;
    const int nn = n0 + pt * 16 + ln;  // this lane's column (point)

    const float* fb  = f + (long long)b * CC * NN;
    const float* sc2 = P + 128;
    const float* sh2 = P + 256;
    const float* sc3 = P + 384;
    const float* sh3 = P + 512;

    #pragma unroll
    for (int mt = 0; mt < 4; ++mt) {
        const int m0 = mh * 64 + mt * 16;
        v8f acc2 = {};
        v8f acc3 = {};
        #pragma unroll
        for (int k = 0; k < CC; k += 4) {
            const int kk = k + 2 * hh;
            // B from gathered/summed LDS tile (w2 path)
            v2f bh;
            bh.x = hs[pt * 16 + ln][kk];
            bh.y = hs[pt * 16 + ln][kk + 1];
            // B from raw feature (w3 shortcut path)
            v2f bf;
            bf.x = fb[(long long)kk * NN + nn];
            bf.y = fb[(long long)(kk + 1) * NN + nn];
            const int m = m0 + ln;
            v2f a2;
            a2.x = w2[m * CC + kk];
            a2.y = w2[m * CC + kk + 1];
            v2f a3;
            a3.x = w3[m * CC + kk];
            a3.y = w3[m * CC + kk + 1];
            acc2 = __builtin_amdgcn_wmma_f32_16x16x4_f32(
                false, a2, false, bh, (short)0, acc2, false, false);
            acc3 = __builtin_amdgcn_wmma_f32_16x16x4_f32(
                false, a3, false, bf, (short)0, acc3, false, false);
        }
        #pragma unroll
        for (int r = 0; r < 8; ++r) {
            const int o = m0 + r + 8 * hh;
            float h2 = fmaxf(acc2[r] * sc2[o] + sh2[o], 0.0f);
            float s3 = fmaxf(acc3[r] * sc3[o] + sh3[o], 0.0f);
            out[((long long)b * DD + o) * NN + nn] = h2 + s3;
        }
    }
}

// ---------------------------------------------------------------------------
extern "C" void kernel_launch(void* const* d_in, const int* in_sizes, int n_in,
                              void* d_out, int out_size, void* d_ws, size_t ws_size,
                              hipStream_t stream)
{
    const float* feature = (const float*)d_in[0];
    const int*   nidx    = (const int*)d_in[1];
    const float* w1      = (const float*)d_in[2];
    const float* g1 = (const float*)d_in[3],  *b1 = (const float*)d_in[4];
    const float* m1 = (const float*)d_in[5],  *v1 = (const float*)d_in[6];
    const float* w2      = (const float*)d_in[7];
    const float* g2 = (const float*)d_in[8],  *b2 = (const float*)d_in[9];
    const float* m2 = (const float*)d_in[10], *v2 = (const float*)d_in[11];
    const float* w3      = (const float*)d_in[12];
    const float* g3 = (const float*)d_in[13], *b3 = (const float*)d_in[14];
    const float* m3 = (const float*)d_in[15], *v3 = (const float*)d_in[16];
    float* out = (float*)d_out;

    float* P  = (float*)d_ws;          // 640 floats of BN params (pad to 1024)
    float* t1 = P + 1024;              // B*N*64 floats = 41.9 MB

    bn_prep<<<1, 128, 0, stream>>>(g1, b1, m1, v1, g2, b2, m2, v2,
                                   g3, b3, m3, v3, P);

    // 128 points per workgroup (8 waves x 16 points)
    stage1<<<(BB * NN) / 128, 256, 0, stream>>>(feature, w1, P, t1);

    // 64 points per workgroup
    stage2<<<(BB * NN) / 64, 256, 0, stream>>>(feature, nidx, t1, w2, w3, P, out);
}